// RandomProjectionQuantizer_11544872092212
// MI455X (gfx1250) — compile-verified
//
#include <hip/hip_runtime.h>
#include <hip/hip_bf16.h>
#include <math.h>

typedef __attribute__((ext_vector_type(2))) float v2f;
typedef __attribute__((ext_vector_type(8))) float v8f;

#define NUM_CODES 8192
#define PROJ_DIM  16
#define KDIM      320           // c * STACK = 80*4
#define ROWS      12000         // b * (t/STACK) = 16*750
#define TILE_M    16

// Single-instruction xor-lane exchange via ds_swizzle_b32 (group-of-32 mode):
// newlane = ((lane & 0x1f) | 0) ^ MASK  -> offset = 0x1f | (MASK << 10)
template <int MASK>
__device__ __forceinline__ float swz_f(float v) {
  return __int_as_float(
      __builtin_amdgcn_ds_swizzle(__float_as_int(v), 0x1f | (MASK << 10)));
}
template <int MASK>
__device__ __forceinline__ int swz_i(int v) {
  return __builtin_amdgcn_ds_swizzle(v, 0x1f | (MASK << 10));
}

// ---------------------------------------------------------------------------
// Kernel 0: l2-normalize codebook rows (8192 x 16) into workspace.
// One thread per row; 4x float4 in/out -> global_load_b128 / store_b128.
// ---------------------------------------------------------------------------
__global__ __launch_bounds__(256)
void rpq_normalize_codebook(const float* __restrict__ cb,
                            float* __restrict__ cbn) {
  int row = blockIdx.x * blockDim.x + threadIdx.x;
  if (row >= NUM_CODES) return;
  const float4* src = (const float4*)(cb + row * PROJ_DIM);
  float4 r0 = src[0], r1 = src[1], r2 = src[2], r3 = src[3];
  float s = r0.x*r0.x + r0.y*r0.y + r0.z*r0.z + r0.w*r0.w
          + r1.x*r1.x + r1.y*r1.y + r1.z*r1.z + r1.w*r1.w
          + r2.x*r2.x + r2.y*r2.y + r2.z*r2.z + r2.w*r2.w
          + r3.x*r3.x + r3.y*r3.y + r3.z*r3.z + r3.w*r3.w;
  float inv = 1.0f / fmaxf(sqrtf(s), 1e-12f);
  r0.x *= inv; r0.y *= inv; r0.z *= inv; r0.w *= inv;
  r1.x *= inv; r1.y *= inv; r1.z *= inv; r1.w *= inv;
  r2.x *= inv; r2.y *= inv; r2.z *= inv; r2.w *= inv;
  r3.x *= inv; r3.y *= inv; r3.z *= inv; r3.w *= inv;
  float4* dst = (float4*)(cbn + row * PROJ_DIM);
  dst[0] = r0; dst[1] = r1; dst[2] = r2; dst[3] = r3;
}

// ---------------------------------------------------------------------------
// Kernel 1: fused  Xp = l2norm(x_tile @ proj);  argmax_n (Xp . cbn[n])
// One wave32 per 16-row tile. All matrix math via v_wmma_f32_16x16x4_f32.
// ---------------------------------------------------------------------------
__global__ __launch_bounds__(32)
void rpq_argmin(const float* __restrict__ x,     // (12000, 320) row-major
                const float* __restrict__ proj,  // (320, 16)    row-major
                const float* __restrict__ cbn,   // (8192, 16)   normalized
                int* __restrict__ out) {         // (12000,)
  __shared__ float lds_xp[TILE_M * PROJ_DIM];    // 1 KB staging tile

  const int lane = threadIdx.x;    // 0..31
  const int n    = lane & 15;      // column / M-row selector
  const int h    = lane >> 4;      // half-wave selector (K split)
  const int row0 = blockIdx.x * TILE_M;

  // ---- Phase 1: acc(16x16) = x_tile(16x320) @ proj(320x16), K in steps of 4.
  // A layout (16x4 f32): lane L -> M = L&15, holds K = k0+2h, k0+2h+1 (contiguous)
  // B layout (4x16 f32): lane L -> N = L&15, holds K = k0+2h, k0+2h+1 (stride 16)
  // Two independent accumulator chains for ILP; merged once at the end.
  v8f acc0 = {}, acc1 = {};
  const float* xrow = x + (size_t)(row0 + n) * KDIM + 2 * h;
  const float* pcol = proj + 2 * h * PROJ_DIM + n;
  #pragma unroll 4
  for (int k0 = 0; k0 < KDIM; k0 += 8) {
    v2f a0 = *(const v2f*)(xrow + k0);          // global_load_b64
    v2f a1 = *(const v2f*)(xrow + k0 + 4);
    v2f b0, b1;
    b0.x = pcol[(k0 + 0) * PROJ_DIM];
    b0.y = pcol[(k0 + 1) * PROJ_DIM];
    b1.x = pcol[(k0 + 4) * PROJ_DIM];
    b1.y = pcol[(k0 + 5) * PROJ_DIM];
    acc0 = __builtin_amdgcn_wmma_f32_16x16x4_f32(
        false, a0, false, b0, (short)0, acc0, false, false);
    acc1 = __builtin_amdgcn_wmma_f32_16x16x4_f32(
        false, a1, false, b1, (short)0, acc1, false, false);
  }
  v8f acc = acc0 + acc1;

  // ---- Row l2-normalize: C/D layout -> VGPR v holds row M = v + 8*h across a
  // 16-lane half, so xor exchanges with masks 1,2,4,8 stay inside the half.
  #pragma unroll
  for (int v = 0; v < 8; ++v) {
    float e = acc[v];
    float s = e * e;
    s += swz_f<1>(s);
    s += swz_f<2>(s);
    s += swz_f<4>(s);
    s += swz_f<8>(s);
    float inv = 1.0f / fmaxf(sqrtf(s), 1e-12f);
    e *= inv;
    lds_xp[(v + 8 * h) * PROJ_DIM + n] = e;     // ds_store_b32
  }
  asm volatile("s_wait_dscnt 0" ::: "memory");

  // ---- Convert D-layout -> A-layout once through LDS; A is loop-invariant.
  v2f a2[4];
  #pragma unroll
  for (int j = 0; j < 4; ++j) {
    // A element (M = n, K = 4*j + 2*h + {0,1})
    a2[j] = *(const v2f*)&lds_xp[n * PROJ_DIM + 4 * j + 2 * h];  // ds_load_b64
  }

  // ---- Phase 2: scan 512 code tiles; scores(16x16) = Xp @ cbn_tile^T (K=16).
  // B element (K = 4*j+2h+{0,1}, N = n) == cbn[(n0+n)][4*j+2h+{0,1}] contiguous.
  // Unrolled x2 -> two independent WMMA/load streams per iteration.
  float best[8];
  int   bidx[8];
  #pragma unroll
  for (int v = 0; v < 8; ++v) { best[v] = -3.0e38f; bidx[v] = 0; }

  #pragma unroll 2
  for (int n0 = 0; n0 < NUM_CODES; n0 += 16) {
    int pf = (n0 + 256 < NUM_CODES) ? (n0 + 256) : (NUM_CODES - 16);
    __builtin_prefetch(cbn + (size_t)pf * PROJ_DIM, 0, 1);   // global_prefetch_b8

    const float* brow = cbn + (size_t)(n0 + n) * PROJ_DIM + 2 * h;
    v8f s = {};
    #pragma unroll
    for (int j = 0; j < 4; ++j) {
      v2f b = *(const v2f*)(brow + 4 * j);      // global_load_b64
      s = __builtin_amdgcn_wmma_f32_16x16x4_f32(
          false, a2[j], false, b, (short)0, s, false, false);
    }
    // Running argmax; candidate index n0+n is monotone per lane, so strict '>'
    // implements argmin-first-occurrence tie-breaking exactly. Branch-free.
    #pragma unroll
    for (int v = 0; v < 8; ++v) {
      bool p  = s[v] > best[v];
      best[v] = p ? s[v] : best[v];
      bidx[v] = p ? (n0 + n) : bidx[v];
    }
  }

  // ---- Cross-lane argmax reduction within each 16-lane half (ties -> min idx)
  #pragma unroll
  for (int v = 0; v < 8; ++v) {
    float bv = best[v];
    int   bi = bidx[v];
    #pragma unroll
    for (int m = 1; m <= 8; m <<= 1) {
      float ov;
      int   oi;
      switch (m) {
        case 1: ov = swz_f<1>(bv); oi = swz_i<1>(bi); break;
        case 2: ov = swz_f<2>(bv); oi = swz_i<2>(bi); break;
        case 4: ov = swz_f<4>(bv); oi = swz_i<4>(bi); break;
        default: ov = swz_f<8>(bv); oi = swz_i<8>(bi); break;
      }
      bool take = (ov > bv) || ((ov == bv) && (oi < bi));
      bv = take ? ov : bv;
      bi = take ? oi : bi;
    }
    if (n == 0) out[row0 + v + 8 * h] = bi;     // rows M = v + 8*h
  }
}

// ---------------------------------------------------------------------------
extern "C" void kernel_launch(void* const* d_in, const int* in_sizes, int n_in,
                              void* d_out, int out_size, void* d_ws, size_t ws_size,
                              hipStream_t stream) {
  const float* x    = (const float*)d_in[0];   // (16,3000,80) == (12000,320)
  const float* proj = (const float*)d_in[1];   // (320,16)
  const float* cb   = (const float*)d_in[2];   // (8192,16)
  int*   out = (int*)d_out;                    // (12000,) int32 argmin indices
  float* cbn = (float*)d_ws;                   // 512 KB normalized codebook

  rpq_normalize_codebook<<<NUM_CODES / 256, 256, 0, stream>>>(cb, cbn);
  rpq_argmin<<<ROWS / TILE_M, 32, 0, stream>>>(x, proj, cbn, out);

  (void)in_sizes; (void)n_in; (void)out_size; (void)ws_size;
}